// DynamicGatedMultiheadAttention_31482110279710
// MI455X (gfx1250) — compile-verified
//
#include <hip/hip_runtime.h>
#include <math.h>

// ---------------------------------------------------------------------------
// DynamicGatedMultiheadAttention, MI455X (gfx1250, wave32, WMMA).
//
// TOP_K == EMBED_DIM, so top_k returns a permutation of ALL rows; the
// gather/batched-matmul/scatter over a full permutation is an identity on
// rows and each DGL layer is exactly  y = x @ W^T + b.  The model reduces to:
//   QKV projections (3x 4096x1024x1024 GEMM) -> MHA (B=4,H=16,L=1024,hd=64)
//   -> output projection (4096x1024x1024 GEMM).
// All matmuls use V_WMMA_F32_16X16X4_F32 (exact fp32, matches reference).
// Staging uses GLOBAL_LOAD_ASYNC_TO_LDS_B128 with LDS double buffering,
// fenced by s_wait_asynccnt + barrier (ASYNCcnt protocol).
// ---------------------------------------------------------------------------

typedef __attribute__((ext_vector_type(2))) float v2f;
typedef __attribute__((ext_vector_type(8))) float v8f;

#define EMB   1024
#define SEQ   1024
#define BATCH 4
#define HEADS 16
#define HDIM  64
#define NROWS (SEQ * BATCH)   // 4096

__device__ __forceinline__ v8f wmma4(v2f a, v2f b, v8f c) {
  // D(16x16,f32) = A(16x4,f32) * B(4x16,f32) + C
  return __builtin_amdgcn_wmma_f32_16x16x4_f32(
      /*neg_a=*/false, a, /*neg_b=*/false, b,
      /*c_mod=*/(short)0, c, /*reuse_a=*/false, /*reuse_b=*/false);
}

// Async copy 16B global -> LDS (per lane). GV addressing: vdst = LDS byte
// address, vaddr = 64-bit global address.  Tracked with ASYNCcnt.
__device__ __forceinline__ void async_b128(void* lds, const float* gptr) {
  const unsigned lds_addr = (unsigned)(size_t)lds;   // low 32b = LDS offset
  asm volatile("global_load_async_to_lds_b128 %0, %1, off"
               :: "v"(lds_addr), "v"(gptr) : "memory");
}
__device__ __forceinline__ void wait_async0() {
  asm volatile("s_wait_asynccnt 0" ::: "memory");
}

// ---------------------------------------------------------------------------
// C[N,M] = A[N,K] * W[M,K]^T + bias[M]      (all fp32, row-major)
// WG = 128 threads = 4 waves (2x2 wave grid).  WG tile: 64 rows x 128 cols.
// Wave tile: 32 rows x 64 cols = 8 accumulators of 16x16.
// K staged async into double-buffered LDS in chunks of 32; rows padded to 36
// floats (16B-aligned stores, conflict-free 16-lane fragment reads).
// ---------------------------------------------------------------------------
__global__ __launch_bounds__(128) void gemm_nt_bias(
    const float* __restrict__ X, const float* __restrict__ W,
    const float* __restrict__ bias, float* __restrict__ Y,
    int N, int K, int M) {
  __shared__ float Xs[2][64][36];     // 18 KB
  __shared__ float Ws[2][128][36];    // 36 KB

  const int tid  = threadIdx.x;
  const int lane = tid & 31;
  const int wave = tid >> 5;       // 0..3
  const int wr   = wave >> 1;      // 0..1  (32-row band)
  const int wc   = wave & 1;       // 0..1  (64-col band)
  const int lm   = lane & 15;
  const int koff = (lane < 16) ? 0 : 2;   // A/B fragment k sub-offset
  const int row0 = blockIdx.y * 64;
  const int col0 = blockIdx.x * 128;

  // Per-thread staging coordinates: 8 threads per 32-float row (float4 each).
  const int sr = tid >> 3;         // 0..15
  const int sc = (tid & 7) * 4;    // 0..28

  v8f acc[2][4];
#pragma unroll
  for (int s = 0; s < 2; ++s)
#pragma unroll
    for (int t = 0; t < 4; ++t)
#pragma unroll
      for (int r = 0; r < 8; ++r) acc[s][t][r] = 0.0f;

  // Issue one chunk of async copies into LDS buffer `buf`.
  auto stage = [&](int buf, int kc) {
#pragma unroll
    for (int i = 0; i < 4; ++i)    // X: 64 rows x 32 cols
      async_b128(&Xs[buf][sr + 16 * i][sc],
                 &X[(size_t)(row0 + sr + 16 * i) * K + kc + sc]);
#pragma unroll
    for (int i = 0; i < 8; ++i)    // W: 128 rows x 32 cols
      async_b128(&Ws[buf][sr + 16 * i][sc],
                 &W[(size_t)(col0 + sr + 16 * i) * K + kc + sc]);
  };

  stage(0, 0);
  for (int kc = 0, ic = 0; kc < K; kc += 32, ++ic) {
    const int buf = ic & 1;
    wait_async0();                 // my async writes to LDS are done
    __syncthreads();               // everyone's writes are visible
    if (kc + 32 < K) stage(buf ^ 1, kc + 32);  // prefetch next chunk

#pragma unroll
    for (int j = 0; j < 8; ++j) {  // 8 k-steps of 4 => 64 WMMA per chunk/wave
      const int k = 4 * j;
      v2f a[2];
#pragma unroll
      for (int s = 0; s < 2; ++s) {
        a[s].x = Xs[buf][32 * wr + 16 * s + lm][k + koff];
        a[s].y = Xs[buf][32 * wr + 16 * s + lm][k + koff + 1];
      }
#pragma unroll
      for (int t = 0; t < 4; ++t) {
        const int ncol = 64 * wc + 16 * t + lm;
        v2f b;
        b.x = Ws[buf][ncol][k + koff];
        b.y = Ws[buf][ncol][k + koff + 1];
        acc[0][t] = wmma4(a[0], b, acc[0][t]);
        acc[1][t] = wmma4(a[1], b, acc[1][t]);
      }
    }
  }

  // Epilogue: C layout -> lane<16 holds rows r, lane>=16 holds rows r+8.
  const int rhi = (lane < 16) ? 0 : 8;
#pragma unroll
  for (int s = 0; s < 2; ++s)
#pragma unroll
    for (int t = 0; t < 4; ++t) {
      const int col = col0 + 64 * wc + 16 * t + lm;
      const float bv = bias[col];
#pragma unroll
      for (int r = 0; r < 8; ++r) {
        const int row = row0 + 32 * wr + 16 * s + r + rhi;
        Y[(size_t)row * M + col] = acc[s][t][r] + bv;
      }
    }
}

// ---------------------------------------------------------------------------
// Flash attention, fp32 WMMA.  Q/K/V/O layout: [L, B, E] flattened as rows
// n = l*B + b of width E; head h occupies columns h*64 .. h*64+63.
// Grid: (L/64, H, B).  WG = 128 threads = 4 waves; wave owns 16 query rows.
// K/V blocks staged async into double-buffered LDS (rows padded to 68 floats:
// 16B-aligned stores; 17*lane mod 16 distinct -> conflict-free frag reads).
// ---------------------------------------------------------------------------
__global__ __launch_bounds__(128) void flash_attn(
    const float* __restrict__ Q, const float* __restrict__ K,
    const float* __restrict__ V, float* __restrict__ O) {
  __shared__ float Kb[2][16][68];
  __shared__ float Vb[2][16][68];
  __shared__ float Ps[4][16][17];     // per-wave P re-layout scratch

  const int tid  = threadIdx.x;
  const int lane = tid & 31;
  const int wave = tid >> 5;          // 0..3
  const int lm   = lane & 15;
  const int koff = (lane < 16) ? 0 : 2;
  const int rhi  = (lane < 16) ? 0 : 8;

  const int b  = blockIdx.z;
  const int h  = blockIdx.y;
  const int q0 = blockIdx.x * 64 + wave * 16;

  // Preload Q fragments for the whole hd=64, pre-scaled by 1/sqrt(hd).
  v2f qa[16];
  {
    const float* qp = Q + ((size_t)(q0 + lm) * BATCH + b) * EMB + h * HDIM;
#pragma unroll
    for (int j = 0; j < 16; ++j) {
      qa[j].x = qp[4 * j + koff] * 0.125f;
      qa[j].y = qp[4 * j + koff + 1] * 0.125f;
    }
  }

  // Per-thread staging coordinates: 16 threads per 64-float row (float4 each).
  const int sr = tid >> 4;           // 0..7
  const int sc = (tid & 15) * 4;     // 0..60
  auto stage = [&](int buf, int kb) {
#pragma unroll
    for (int i = 0; i < 2; ++i) {
      const size_t base =
          ((size_t)(kb + sr + 8 * i) * BATCH + b) * EMB + h * HDIM + sc;
      async_b128(&Kb[buf][sr + 8 * i][sc], &K[base]);
      async_b128(&Vb[buf][sr + 8 * i][sc], &V[base]);
    }
  };

  v8f o[4];
#pragma unroll
  for (int t = 0; t < 4; ++t)
#pragma unroll
    for (int r = 0; r < 8; ++r) o[t][r] = 0.0f;
  float mrow[8], lrow[8];
#pragma unroll
  for (int r = 0; r < 8; ++r) { mrow[r] = -1e30f; lrow[r] = 0.0f; }

  stage(0, 0);
  for (int kb = 0; kb < SEQ; kb += 16) {
    const int buf = (kb >> 4) & 1;
    wait_async0();
    __syncthreads();
    if (kb + 16 < SEQ) stage(buf ^ 1, kb + 16);

    // S(16x16) = Qtile(16x64) * Kblk(16x64)^T  -> 16 WMMA k-steps
    v8f s;
#pragma unroll
    for (int r = 0; r < 8; ++r) s[r] = 0.0f;
#pragma unroll
    for (int j = 0; j < 16; ++j) {
      v2f bb;                          // B[k=d, n=keyIdx] = Kb[n][d]
      bb.x = Kb[buf][lm][4 * j + koff];
      bb.y = Kb[buf][lm][4 * j + koff + 1];
      s = wmma4(qa[j], bb, s);
    }

    // Online softmax (rows live in 16-lane groups; xor 1/2/4/8 stays in-group)
    float alpha[8];
#pragma unroll
    for (int r = 0; r < 8; ++r) {
      float mx = s[r];
      mx = fmaxf(mx, __shfl_xor(mx, 1, 32));
      mx = fmaxf(mx, __shfl_xor(mx, 2, 32));
      mx = fmaxf(mx, __shfl_xor(mx, 4, 32));
      mx = fmaxf(mx, __shfl_xor(mx, 8, 32));
      const float mnew = fmaxf(mrow[r], mx);
      alpha[r] = __expf(mrow[r] - mnew);
      mrow[r]  = mnew;
      const float p = __expf(s[r] - mnew);
      s[r] = p;
      float sum = p;
      sum += __shfl_xor(sum, 1, 32);
      sum += __shfl_xor(sum, 2, 32);
      sum += __shfl_xor(sum, 4, 32);
      sum += __shfl_xor(sum, 8, 32);
      lrow[r] = lrow[r] * alpha[r] + sum;
    }
#pragma unroll
    for (int t = 0; t < 4; ++t)
#pragma unroll
      for (int r = 0; r < 8; ++r) o[t][r] *= alpha[r];

    // Re-layout P: C-tile layout -> A-fragment layout via wave-private LDS.
    // DS ops are in-order within a wave, so no block barrier is needed here.
#pragma unroll
    for (int r = 0; r < 8; ++r) Ps[wave][r + rhi][lm] = s[r];
    v2f pa[4];
#pragma unroll
    for (int j = 0; j < 4; ++j) {
      pa[j].x = Ps[wave][lm][4 * j + koff];
      pa[j].y = Ps[wave][lm][4 * j + koff + 1];
    }

    // O(16x64) += P(16x16) * Vblk(16x64)   -> 4x4 WMMA
#pragma unroll
    for (int t = 0; t < 4; ++t) {
#pragma unroll
      for (int j = 0; j < 4; ++j) {
        v2f bv;                        // B[k=keyIdx, n=d] = Vb[k][16t+n]
        bv.x = Vb[buf][4 * j + koff][16 * t + lm];
        bv.y = Vb[buf][4 * j + koff + 1][16 * t + lm];
        o[t] = wmma4(pa[j], bv, o[t]);
      }
    }
  }

  // Normalize and store O in [L,B,E] layout (feeds the output projection).
#pragma unroll
  for (int r = 0; r < 8; ++r) {
    const float inv = 1.0f / lrow[r];
    const int l = q0 + r + rhi;
#pragma unroll
    for (int t = 0; t < 4; ++t) {
      O[((size_t)l * BATCH + b) * EMB + h * HDIM + 16 * t + lm] = o[t][r] * inv;
    }
  }
}

// ---------------------------------------------------------------------------
extern "C" void kernel_launch(void* const* d_in, const int* in_sizes, int n_in,
                              void* d_out, int out_size, void* d_ws, size_t ws_size,
                              hipStream_t stream) {
  const float* query = (const float*)d_in[0];
  const float* key   = (const float*)d_in[1];
  const float* value = (const float*)d_in[2];
  const float* ipw   = (const float*)d_in[3];   // [3E, E]
  const float* ipb   = (const float*)d_in[4];   // [3E]
  const float* outw  = (const float*)d_in[5];   // [E, E]
  const float* outb  = (const float*)d_in[6];   // [E]
  // d_in[7..18]: layernorm + gate-MLP params. Mathematically dead: TOP_K == E
  // makes the gather/scatter a row permutation identity, so they are unused.

  float* out = (float*)d_out;
  float* ws  = (float*)d_ws;
  const size_t MAT = (size_t)NROWS * EMB;       // 4M floats = 16 MB
  float* Qb = ws;
  float* Kb = ws + MAT;
  float* Vb = ws + 2 * MAT;
  float* Ob = ws + 3 * MAT;

  const dim3 gridG(EMB / 128, NROWS / 64);      // (8, 64)
  gemm_nt_bias<<<gridG, 128, 0, stream>>>(query, ipw,                         ipb,           Qb, NROWS, EMB, EMB);
  gemm_nt_bias<<<gridG, 128, 0, stream>>>(key,   ipw + (size_t)EMB * EMB,     ipb + EMB,     Kb, NROWS, EMB, EMB);
  gemm_nt_bias<<<gridG, 128, 0, stream>>>(value, ipw + 2 * (size_t)EMB * EMB, ipb + 2 * EMB, Vb, NROWS, EMB, EMB);

  const dim3 gridA(SEQ / 64, HEADS, BATCH);     // (16, 16, 4)
  flash_attn<<<gridA, 128, 0, stream>>>(Qb, Kb, Vb, Ob);

  gemm_nt_bias<<<gridG, 128, 0, stream>>>(Ob, outw, outb, out, NROWS, EMB, EMB);
}